// SPTransformer_30210799960554
// MI455X (gfx1250) — compile-verified
//
#include <hip/hip_runtime.h>

#define SB     784
#define NB     785
#define NBATCH 8
#define NCH    12
#define DDIM   768
#define HH     28
#define KPN    84
#define SELN   42
#define KHID   512

typedef __attribute__((ext_vector_type(2)))  float    v2f;
typedef __attribute__((ext_vector_type(8)))  float    v8f;

// ---- workspace layout (float offsets) ----
#define WS_NEWSCORE 0                               // B*C*S = 75264
#define WS_COUNT    (WS_NEWSCORE + NBATCH*NCH*SB)   // B*S   = 6272
#define WS_G        (WS_COUNT + NBATCH*SB)          // B*512 = 4096
#define WS_PWA      (WS_G + NBATCH*KHID)            // B
#define WS_STRUCT   (WS_PWA + NBATCH)               // B*768 (16B aligned: 85640%4==0)
#define WS_ANCHOR   (WS_STRUCT + NBATCH*DDIM)       // B ints
#define WS_PIDX     (WS_ANCHOR + NBATCH)            // B*42 ints

__global__ void k_zero(float* p, int n) {
    int i = blockIdx.x * 256 + threadIdx.x;
    if (i < n) p[i] = 0.0f;
}

// Per (b,c): score = x[b,c,0,1:], exact stable top-84 via rank, new_score, count scatter.
__global__ void k_topk(const float* __restrict__ x,
                       float* __restrict__ new_score,
                       float* __restrict__ count) {
    const int bc = blockIdx.x;          // b*NCH + c
    const int b  = bc / NCH;
    const float* row = x + (size_t)bc * NB * NB + 1;   // [b,c,0,1+s]
    __shared__ float s[SB];
    for (int i = threadIdx.x; i < SB; i += 256) s[i] = row[i];
    __syncthreads();
    for (int i = threadIdx.x; i < SB; i += 256) {
        const float v = s[i];
        int rank = 0;
        for (int j = 0; j < SB; ++j) {
            const float w = s[j];
            rank += (w > v) || ((w == v) && (j < i));
        }
        const bool sel = rank < KPN;
        new_score[(size_t)bc * SB + i] = sel ? v : v * 0.7f;
        if (sel) atomicAdd(&count[b * SB + i], 1.0f);
    }
}

// Per batch: pw, binary, anchor argmax, Pp/Pn, u = pw.SI, g = v*(v>0?Pp:Pn)
__global__ void k_stats(const float* __restrict__ ns_all,
                        const float* __restrict__ w1,
                        float* __restrict__ g_out,
                        float* __restrict__ pwa_out,
                        int* __restrict__ anchor_out) {
    const int b = blockIdx.x;
    const int t = threadIdx.x;
    __shared__ float pw[SB];
    __shared__ float red[256];
    __shared__ int   redi[256];
    __shared__ float s_mean, s_pp, s_pn, s_u0, s_u1;
    __shared__ int   s_anchor;

    const float* ns = ns_all + (size_t)b * NCH * SB;
    for (int i = t; i < SB; i += 256) {
        float s = 0.f;
        #pragma unroll
        for (int c = 0; c < NCH; ++c) s += ns[c * SB + i];
        pw[i] = s * (1.0f / NCH);
    }
    __syncthreads();

    float acc = 0.f;
    for (int i = t; i < SB; i += 256) acc += pw[i];
    red[t] = acc; __syncthreads();
    for (int off = 128; off > 0; off >>= 1) { if (t < off) red[t] += red[t + off]; __syncthreads(); }
    if (t == 0) s_mean = red[0] * (1.0f / SB);
    __syncthreads();
    const float meanpw = s_mean;

    float bv = -__builtin_inff(); int bi = 0x7fffffff;
    for (int i = t; i < SB; i += 256) {
        const float val = (pw[i] > meanpw) ? pw[i] : 0.0f;   // == binary*pw
        if (val > bv || (val == bv && i < bi)) { bv = val; bi = i; }
    }
    red[t] = bv; redi[t] = bi; __syncthreads();
    for (int off = 128; off > 0; off >>= 1) {
        if (t < off) {
            const float ov = red[t + off]; const int oi = redi[t + off];
            if (ov > red[t] || (ov == red[t] && oi < redi[t])) { red[t] = ov; redi[t] = oi; }
        }
        __syncthreads();
    }
    if (t == 0) s_anchor = redi[0];
    __syncthreads();
    const int anchor = s_anchor;
    const int ar = anchor / HH, acol = anchor % HH;

    float pp = 0.f, pn = 0.f, u0 = 0.f, u1 = 0.f;
    for (int i = t; i < SB; i += 256) {
        const float p = pw[i];
        if (p > 0.f) pp += p * p; else pn += p * p;
        const float dy = (float)(i / HH - ar) * (1.0f / HH);
        const float dx = (float)(i % HH - acol) * (1.0f / HH);
        const float dist = sqrtf(dy * dy + dx * dx);
        const float ang  = (atan2f(dx, dy) * (1.0f / 3.14159265358979323846f) + 1.0f) * 0.5f;
        u0 += p * dist;
        u1 += p * ang;
    }
    red[t] = pp; __syncthreads();
    for (int off = 128; off > 0; off >>= 1) { if (t < off) red[t] += red[t + off]; __syncthreads(); }
    if (!t) s_pp = red[0]; __syncthreads();
    red[t] = pn; __syncthreads();
    for (int off = 128; off > 0; off >>= 1) { if (t < off) red[t] += red[t + off]; __syncthreads(); }
    if (!t) s_pn = red[0]; __syncthreads();
    red[t] = u0; __syncthreads();
    for (int off = 128; off > 0; off >>= 1) { if (t < off) red[t] += red[t + off]; __syncthreads(); }
    if (!t) s_u0 = red[0]; __syncthreads();
    red[t] = u1; __syncthreads();
    for (int off = 128; off > 0; off >>= 1) { if (t < off) red[t] += red[t + off]; __syncthreads(); }
    if (!t) s_u1 = red[0]; __syncthreads();

    const float Pp = s_pp, Pn = s_pn, U0 = s_u0, U1 = s_u1;
    for (int k = t; k < KHID; k += 256) {
        const float v = U0 * w1[k] + U1 * w1[KHID + k];      // w1 is (2,512)
        g_out[(size_t)b * KHID + k] = v * (v > 0.f ? Pp : Pn);
    }
    if (t == 0) { pwa_out[b] = pw[anchor]; anchor_out[b] = anchor; }
}

// One wave per 16-wide N tile: R = Gpad(16x512) @ W2(512x768); struct = leaky(pw_anchor * R).
// A and B tiles staged in LDS: branch-free ds_load in the WMMA loop (EXEC stays all-1s).
__global__ void __launch_bounds__(32)
k_gemm_wmma(const float* __restrict__ g,
            const float* __restrict__ w2,
            const float* __restrict__ pwa,
            float* __restrict__ structv) {
    __shared__ float As[16 * KHID];     // 32 KB: rows 0-7 = G, rows 8-15 = 0
    __shared__ float Bs[KHID * 16];     // 32 KB: W2[k][n0+0..15], row-major k x 16
    const int n0   = blockIdx.x * 16;
    const int lane = threadIdx.x;
    const int m    = lane & 15;
    const int kh   = lane >> 4;         // lane half selects K pair

    // Fill A: 8*512 floats via b128, then zero pad rows.
    {
        const float4* gsrc = (const float4*)g;
        float4*       adst = (float4*)As;
        #pragma unroll
        for (int i = 0; i < (NBATCH * KHID / 4) / 32; ++i)
            adst[lane + 32 * i] = gsrc[lane + 32 * i];
        float4 z; z.x = z.y = z.z = z.w = 0.f;
        float4* zdst = (float4*)(As + NBATCH * KHID);
        #pragma unroll
        for (int i = 0; i < (NBATCH * KHID / 4) / 32; ++i)
            zdst[lane + 32 * i] = z;
    }
    // Fill B: each lane copies rows k = lane, lane+32, ... (16 contiguous floats each).
    for (int k = lane; k < KHID; k += 32) {
        const float4* src = (const float4*)(w2 + (size_t)k * DDIM + n0);
        float4*       dst = (float4*)(Bs + k * 16);
        dst[0] = src[0]; dst[1] = src[1]; dst[2] = src[2]; dst[3] = src[3];
    }
    __syncthreads();

    v8f acc = {};
    #pragma unroll 8
    for (int k0 = 0; k0 < KHID; k0 += 4) {
        const int ka = k0 + 2 * kh;
        // A 16x4 f32: lanes0-15 V0=K0,V1=K1; lanes16-31 V0=K2,V1=K3 (M = lane&15)
        const float2 ap = *(const float2*)&As[m * KHID + ka];   // ds_load_b64
        v2f a, bb;
        a[0] = ap.x; a[1] = ap.y;
        // B 4x16 f32: symmetric layout, N = lane&15
        bb[0] = Bs[ka * 16 + m];
        bb[1] = Bs[(ka + 1) * 16 + m];
        acc = __builtin_amdgcn_wmma_f32_16x16x4_f32(false, a, false, bb, (short)0, acc,
                                                    false, false);
    }

    // C/D: lanes0-15 VGPR r -> M=r (we only need M<8), N = lane
    if (lane < 16) {
        #pragma unroll
        for (int r = 0; r < NBATCH; ++r) {
            const float o = pwa[r] * acc[r];
            structv[(size_t)r * DDIM + n0 + m] = (o > 0.f) ? o : 0.2f * o;
        }
    }
}

// hs = hidden_states; hs[:,0,:] += struct[b]; bulk b128 copy
__global__ void k_hs(const float* __restrict__ hidden,
                     const float* __restrict__ structv,
                     float* __restrict__ out) {
    const size_t idx = (size_t)blockIdx.x * 256 + threadIdx.x;      // float4 index
    const size_t total = (size_t)NBATCH * NB * DDIM / 4;
    if (idx >= total) return;
    float4 v = ((const float4*)hidden)[idx];
    const size_t f   = idx * 4;
    const size_t per = (size_t)NB * DDIM;
    const int    b   = (int)(f / per);
    const size_t r   = f % per;
    if (r < DDIM) {
        const float4 sv = ((const float4*)(structv + (size_t)b * DDIM))[r >> 2];
        v.x += sv.x; v.y += sv.y; v.z += sv.z; v.w += sv.w;
    }
    ((float4*)out)[idx] = v;
}

// 3x3 [[1,2,1],[2,4,2],[1,2,1]] SAME conv on count, then stable descending ranks -> patch_idx
__global__ void k_conv_rank(const float* __restrict__ count,
                            int* __restrict__ patch_idx) {
    const int b = blockIdx.x;
    __shared__ float cnt[SB];
    __shared__ float cv[SB];
    for (int i = threadIdx.x; i < SB; i += 256) cnt[i] = count[b * SB + i];
    __syncthreads();
    const float wk[3][3] = {{1.f,2.f,1.f},{2.f,4.f,2.f},{1.f,2.f,1.f}};
    for (int i = threadIdx.x; i < SB; i += 256) {
        const int r = i / HH, c = i % HH;
        float s = 0.f;
        #pragma unroll
        for (int dr = -1; dr <= 1; ++dr)
            #pragma unroll
            for (int dc = -1; dc <= 1; ++dc) {
                const int rr = r + dr, cc = c + dc;
                if (rr >= 0 && rr < HH && cc >= 0 && cc < HH)
                    s += wk[dr + 1][dc + 1] * cnt[rr * HH + cc];
            }
        cv[i] = s;
    }
    __syncthreads();
    for (int i = threadIdx.x; i < SB; i += 256) {
        const float v = cv[i];
        int rank = 0;
        for (int j = 0; j < SB; ++j) {
            const float w = cv[j];
            rank += (w > v) || ((w == v) && (j < i));   // stable descending order
        }
        if (rank < SELN) patch_idx[b * SELN + rank] = i + 1;
    }
}

// selected[b,r,:] = hs[b, patch_idx[b,r], :]
__global__ void k_gather(const float* __restrict__ hs,
                         const int* __restrict__ patch_idx,
                         float* __restrict__ selected) {
    const int br = blockIdx.x;
    const int b  = br / SELN;
    const int r  = br % SELN;
    const int p  = patch_idx[b * SELN + r];
    const float4* src = (const float4*)(hs + ((size_t)b * NB + p) * DDIM);
    float4*       dst = (float4*)(selected + ((size_t)b * SELN + r) * DDIM);
    dst[threadIdx.x] = src[threadIdx.x];
}

extern "C" void kernel_launch(void* const* d_in, const int* in_sizes, int n_in,
                              void* d_out, int out_size, void* d_ws, size_t ws_size,
                              hipStream_t stream) {
    const float* hidden = (const float*)d_in[0];   // (8,785,768)
    const float* x      = (const float*)d_in[1];   // (8,12,785,785)
    const float* w1     = (const float*)d_in[2];   // (2,512)
    const float* w2     = (const float*)d_in[3];   // (512,768)
    (void)in_sizes; (void)n_in; (void)out_size; (void)ws_size;

    float* wsf = (float*)d_ws;
    float* out = (float*)d_out;

    k_zero<<<(NBATCH * SB + 255) / 256, 256, 0, stream>>>(wsf + WS_COUNT, NBATCH * SB);
    k_topk<<<NBATCH * NCH, 256, 0, stream>>>(x, wsf + WS_NEWSCORE, wsf + WS_COUNT);
    k_stats<<<NBATCH, 256, 0, stream>>>(wsf + WS_NEWSCORE, w1, wsf + WS_G,
                                        wsf + WS_PWA, (int*)(wsf + WS_ANCHOR));
    k_gemm_wmma<<<DDIM / 16, 32, 0, stream>>>(wsf + WS_G, w2, wsf + WS_PWA,
                                              wsf + WS_STRUCT);
    const int hs4 = NBATCH * NB * DDIM / 4;
    k_hs<<<(hs4 + 255) / 256, 256, 0, stream>>>(hidden, wsf + WS_STRUCT, out);
    k_conv_rank<<<NBATCH, 256, 0, stream>>>(wsf + WS_COUNT, (int*)(wsf + WS_PIDX));
    k_gather<<<NBATCH * SELN, DDIM / 4, 0, stream>>>(out, (int*)(wsf + WS_PIDX),
                                                     out + (size_t)NBATCH * NB * DDIM);
}